// SubBiAct_83408264888357
// MI455X (gfx1250) — compile-verified
//
#include <hip/hip_runtime.h>
#include <hip/hip_bf16.h>

// SubBiAct: out = clamp(x,0,1); hoyer = sum(out^2)/sum(out);
// result = (out >= |scores[c]| * hoyer) ? 1 : 0
// Shapes fixed by reference: N=64, C=128, H=64, W=64  (HW = 4096 = 1<<12)

typedef float v2f __attribute__((ext_vector_type(2)));
typedef float v4f __attribute__((ext_vector_type(4)));
typedef float v8f __attribute__((ext_vector_type(8)));

#define CCH 128           // channels (power of two)
#define HW_LOG2 12        // H*W = 4096
#define P1_BLOCKS 1024
#define P1_THREADS 256

// ---------------------------------------------------------------------------
// Wave-level reduction of two independent partial sums (s1, s2) using a single
// V_WMMA_F32_16X16X4_F32.
//   A (16x4 f32): lane m (<16) supplies A[m][0]=s1, A[m][1]=s2;
//                 lane m+16 supplies A[m][2]=s1, A[m][3]=s2.
//   B (4x16 f32): B[0][n]=B[2][n]=(n<8), B[1][n]=B[3][n]=(n>=8).
//   => D[m][n<8] = s1[m]+s1[m+16],  D[m][n>=8] = s2[m]+s2[m+16].
// Summing the 8 C/D VGPRs in-lane gives t[lane]; t + shfl_xor(t,16) yields
// u[lane] = total_s1 if (lane&15)<8 else total_s2, on every lane.
// ---------------------------------------------------------------------------
__device__ __forceinline__ float wave_wmma_reduce2(float s1, float s2) {
    unsigned lane = threadIdx.x & 31u;
    v2f a; a.x = s1; a.y = s2;
    float bx = ((lane & 15u) < 8u) ? 1.0f : 0.0f;
    v2f b; b.x = bx; b.y = 1.0f - bx;
    v8f c = {};
    c = __builtin_amdgcn_wmma_f32_16x16x4_f32(
        /*neg_a=*/false, a, /*neg_b=*/false, b,
        /*c_mod=*/(short)0, c, /*reuse_a=*/false, /*reuse_b=*/false);
    float t = c[0] + c[1] + c[2] + c[3] + c[4] + c[5] + c[6] + c[7];
    float u = t + __shfl_xor(t, 16, 32);
    return u; // (lane&15)<8 -> total s1 ; else total s2
}

__device__ __forceinline__ float sat01(float x) {
    return fminf(fmaxf(x, 0.0f), 1.0f); // relu + (x>=1 -> 1), MIN=0, MAX=1
}

// Pass 1: per-block partial (sum(out), sum(out^2)) -> part[2*block .. +1]
__global__ __launch_bounds__(P1_THREADS)
void subbiact_reduce(const float* __restrict__ in, float* __restrict__ part,
                     long long n) {
    const v4f* in4 = reinterpret_cast<const v4f*>(in);
    long long nvec   = n >> 2;
    long long stride = (long long)gridDim.x * blockDim.x;
    long long tid    = (long long)blockIdx.x * blockDim.x + threadIdx.x;

    float s1 = 0.0f, s2 = 0.0f;
    for (long long i = tid; i < nvec; i += stride) {
        v4f x = in4[i];
        float o0 = sat01(x.x), o1 = sat01(x.y), o2 = sat01(x.z), o3 = sat01(x.w);
        s1 += (o0 + o1) + (o2 + o3);
        s2 += (o0 * o0 + o1 * o1) + (o2 * o2 + o3 * o3);
    }
    // scalar tail (empty for the reference shape; kept for generality)
    for (long long e = (nvec << 2) + tid; e < n; e += stride) {
        float o = sat01(in[e]);
        s1 += o; s2 += o * o;
    }

    float u = wave_wmma_reduce2(s1, s2);

    __shared__ float lds[2 * (P1_THREADS / 32)];
    unsigned lane = threadIdx.x & 31u;
    unsigned wave = threadIdx.x >> 5;
    if (lane == 0) lds[2 * wave]     = u; // wave total s1
    if (lane == 8) lds[2 * wave + 1] = u; // wave total s2
    __syncthreads();
    if (threadIdx.x == 0) {
        float a = 0.0f, b = 0.0f;
        #pragma unroll
        for (int w = 0; w < P1_THREADS / 32; ++w) { a += lds[2 * w]; b += lds[2 * w + 1]; }
        part[2 * blockIdx.x]     = a;
        part[2 * blockIdx.x + 1] = b;
    }
}

// Pass 2 (one wave): reduce block partials, compute hoyer, emit per-channel thr
__global__ void subbiact_finalize(const float* __restrict__ scores,
                                  const float* __restrict__ part,
                                  float* __restrict__ thr, int npart) {
    unsigned lane = threadIdx.x; // blockDim.x == 32
    float s1 = 0.0f, s2 = 0.0f;
    for (int i = (int)lane; i < npart; i += 32) {
        s1 += part[2 * i];
        s2 += part[2 * i + 1];
    }
    float u  = wave_wmma_reduce2(s1, s2);
    float t1 = __shfl(u, 0, 32); // total sum(out)
    float t2 = __shfl(u, 8, 32); // total sum(out^2)
    float hoyer = t2 / t1;
    for (int c = (int)lane; c < CCH; c += 32)
        thr[c] = fabsf(scores[c]) * hoyer;
}

// Pass 3: binarize. Input should hit in the 192MB L2; NT stores for the
// write-once output so it doesn't evict the resident input.
__global__ __launch_bounds__(256)
void subbiact_binarize(const float* __restrict__ in, const float* __restrict__ thr,
                       float* __restrict__ out, long long n) {
    const v4f* in4  = reinterpret_cast<const v4f*>(in);
    v4f*       out4 = reinterpret_cast<v4f*>(out);
    long long nvec   = n >> 2;
    long long stride = (long long)gridDim.x * blockDim.x;
    long long tid    = (long long)blockIdx.x * blockDim.x + threadIdx.x;

    for (long long i = tid; i < nvec; i += stride) {
        v4f x = in4[i];
        // 4 consecutive elements share a channel: c = (4i >> 12) & (C-1)
        int cch = (int)((i >> (HW_LOG2 - 2)) & (CCH - 1));
        float th = thr[cch];
        v4f r;
        r.x = (sat01(x.x) >= th) ? 1.0f : 0.0f;
        r.y = (sat01(x.y) >= th) ? 1.0f : 0.0f;
        r.z = (sat01(x.z) >= th) ? 1.0f : 0.0f;
        r.w = (sat01(x.w) >= th) ? 1.0f : 0.0f;
        __builtin_nontemporal_store(r, &out4[i]);
    }
    for (long long e = (nvec << 2) + tid; e < n; e += stride) {
        int cch = (int)((e >> HW_LOG2) & (CCH - 1));
        out[e] = (sat01(in[e]) >= thr[cch]) ? 1.0f : 0.0f;
    }
}

extern "C" void kernel_launch(void* const* d_in, const int* in_sizes, int n_in,
                              void* d_out, int out_size, void* d_ws, size_t ws_size,
                              hipStream_t stream) {
    const float* in     = (const float*)d_in[0];
    const float* scores = (const float*)d_in[1];
    (void)n_in; (void)ws_size; // d_in[2] = epoch (unused by the math)

    long long n = (long long)in_sizes[0];

    float* part = (float*)d_ws;           // 2 * P1_BLOCKS floats of partials
    float* thr  = part + 2 * P1_BLOCKS;   // CCH per-channel thresholds
    float* out  = (float*)d_out;

    subbiact_reduce  <<<P1_BLOCKS, P1_THREADS, 0, stream>>>(in, part, n);
    subbiact_finalize<<<1, 32, 0, stream>>>(scores, part, thr, P1_BLOCKS);
    subbiact_binarize<<<2048, 256, 0, stream>>>(in, thr, out, n);
}